// GraphTransformer_16904991277161
// MI455X (gfx1250) — compile-verified
//
#include <hip/hip_runtime.h>
#include <hip/hip_bf16.h>

#define BB   16
#define NN   1024
#define DIN  128
#define DH   256
#define NP1  1025

typedef __attribute__((ext_vector_type(2))) float v2f;
typedef __attribute__((ext_vector_type(8))) float v8f;

// ---------------------------------------------------------------------------
// Kernel 1: tiny MLP -> lut[0..5] = (relu(codebook@W1 + b1) @ W2 + b2), lut[6]=0
// Single wave (32 lanes), V_WMMA_F32_16X16X4_F32 (exact f32 math).
// A = codebook padded to 16 rows; A fragment hoisted into registers (64 VGPRs,
// fine for one resident wave) and reused across all 16 N-tiles. Padding rows
// are produced branch-free via a 0/1 lane mask (no EXEC save/restore).
// ---------------------------------------------------------------------------
__global__ void mlp_lut_kernel(const float* __restrict__ codebook,
                               const float* __restrict__ W1,
                               const float* __restrict__ b1,
                               const float* __restrict__ W2,
                               const float* __restrict__ b2,
                               float* __restrict__ lut) {
    const int lane  = threadIdx.x;      // 0..31, one wave32
    const int half  = lane >> 4;        // 0: lanes 0-15, 1: lanes 16-31
    const int lhalf = lane & 15;

    __shared__ float red[8][32];

#if defined(__gfx1250__) && __has_builtin(__builtin_amdgcn_wmma_f32_16x16x4_f32)
    // ---- WMMA path -------------------------------------------------------
    const int   m     = lhalf;                       // A row held by this lane
    const float amask = (m < 6) ? 1.f : 0.f;         // zero-pad rows 6..15
    const int   mrow  = (m < 6) ? m : 0;             // clamped (always valid) row

    // Preload full A fragment: K layout per ISA — lanes 0-15 hold K={0,1},
    // lanes 16-31 hold K={2,3} of each 4-wide K slab.
    v2f afrag[DIN / 4];
#pragma unroll
    for (int k4 = 0; k4 < DIN / 4; ++k4) {
        const int k0 = k4 * 4 + (half ? 2 : 0);
        afrag[k4].x = amask * codebook[mrow * DIN + k0];
        afrag[k4].y = amask * codebook[mrow * DIN + k0 + 1];
    }

    float partial[8];
#pragma unroll
    for (int v = 0; v < 8; ++v) partial[v] = 0.f;

    for (int nt = 0; nt < DH / 16; ++nt) {
        const int ncol = nt * 16 + lhalf;            // B/C column held by lane
        v8f acc = {0.f, 0.f, 0.f, 0.f, 0.f, 0.f, 0.f, 0.f};
#pragma unroll
        for (int k4 = 0; k4 < DIN / 4; ++k4) {
            const int k0 = k4 * 4 + (half ? 2 : 0);
            v2f bf;
            bf.x = W1[k0 * DH + ncol];
            bf.y = W1[(k0 + 1) * DH + ncol];
            acc = __builtin_amdgcn_wmma_f32_16x16x4_f32(
                false, afrag[k4], false, bf, (short)0, acc, false, false);
        }
        // acc[v] = H_pre[m = v + 8*half][ncol]; fold bias + relu + W2 dot
        const float bias = b1[ncol];
        const float w2   = W2[ncol];
#pragma unroll
        for (int v = 0; v < 8; ++v) {
            float h = acc[v] + bias;
            h = h > 0.f ? h : 0.f;
            partial[v] += h * w2;
        }
    }
    // lanes 16-31 hold rows 8..15 (all padding) -> contribute zero
#pragma unroll
    for (int v = 0; v < 8; ++v)
        red[v][lane] = (half == 0) ? partial[v] : 0.f;
#else
    // ---- scalar fallback (host pass / missing builtin) -------------------
    float pm[8];
#pragma unroll
    for (int v = 0; v < 8; ++v) pm[v] = 0.f;
    for (int n = lane; n < DH; n += 32) {
        const float bias = b1[n];
        const float w2   = W2[n];
        for (int mm = 0; mm < 6; ++mm) {
            float s = bias;
            for (int k = 0; k < DIN; ++k)
                s += codebook[mm * DIN + k] * W1[k * DH + n];
            s = s > 0.f ? s : 0.f;
            pm[mm] += s * w2;
        }
    }
#pragma unroll
    for (int v = 0; v < 8; ++v) red[v][lane] = (v < 6) ? pm[v] : 0.f;
#endif

    __syncthreads();
    if (lane < 6) {
        float s = b2[0];
        for (int l = 0; l < 32; ++l) s += red[lane][l];
        lut[lane] = s;
    }
    if (lane == 6) lut[6] = 0.f;
}

// ---------------------------------------------------------------------------
// Kernel 2: input_X = concat(tst_token, features) along rows. float4 streams.
// ---------------------------------------------------------------------------
__global__ void build_X_kernel(const float* __restrict__ feat,
                               const float* __restrict__ tst,
                               float* __restrict__ outX) {
    const int per_row = DIN / 4;          // 32 float4 per row
    const int per_b   = NP1 * per_row;
    const int idx = blockIdx.x * blockDim.x + threadIdx.x;
    if (idx >= BB * per_b) return;
    const int b   = idx / per_b;
    const int rem = idx % per_b;
    const int r   = rem / per_row;
    const int c   = rem % per_row;
    float4 v;
    if (r == 0) v = ((const float4*)tst)[c];
    else        v = ((const float4*)feat)[((size_t)b * NN + (r - 1)) * per_row + c];
    ((float4*)outX)[idx] = v;
}

// ---------------------------------------------------------------------------
// Kernel 3: input_MASK = concat(ones, mask) as floats.
// ---------------------------------------------------------------------------
__global__ void build_mask_kernel(const unsigned char* __restrict__ mask,
                                  float* __restrict__ outM) {
    const int idx = blockIdx.x * blockDim.x + threadIdx.x;
    if (idx >= BB * NP1) return;
    const int b = idx / NP1;
    const int r = idx % NP1;
    outM[idx] = (r == 0) ? 1.f : (mask[b * NN + r - 1] ? 1.f : 0.f);
}

// ---------------------------------------------------------------------------
// Kernel 4: bias_mat[b][i][j] = lut[sel(D[b][i][j])], one block per output row.
// Interior reads dm rows contiguously; each b32 load/store instruction covers
// a fully-contiguous 128B wave32 span (coalesced). Border elements use the
// row0 "+1 unless 9999" rule. LUT resolves to a v_cndmask chain over 6 SGPRs.
// ---------------------------------------------------------------------------
__global__ void build_bias_kernel(const int* __restrict__ dm,
                                  const float* __restrict__ lut,
                                  float* __restrict__ outB) {
    const int b = blockIdx.x / NP1;
    const int i = blockIdx.x % NP1;
    const int* __restrict__ dmb = dm + (size_t)b * NN * NN;
    const float l0 = lut[0], l1 = lut[1], l2 = lut[2];
    const float l3 = lut[3], l4 = lut[4], l5 = lut[5];
    float* __restrict__ row = outB + (size_t)blockIdx.x * NP1;

    // Source row + whether the "+1 unless 9999" bump applies (i==0 border row)
    const int* __restrict__ srow = (i == 0) ? dmb : (dmb + (size_t)(i - 1) * NN);
    const bool bump = (i == 0);

    for (int j = threadIdx.x; j < NP1; j += blockDim.x) {
        int d;
        if (j == 0) {
            if (i == 0) d = 0;
            else { d = dmb[i - 1]; d = (d == 9999) ? 9999 : d + 1; }
        } else {
            d = srow[j - 1];
            if (bump) d = (d == 9999) ? 9999 : d + 1;
        }
        // sel: 9999 -> lut[5]; d in 0..4 -> lut[d]; else -> lut[6] == 0
        float v = (d == 9999) ? l5
                : (d == 0) ? l0
                : (d == 1) ? l1
                : (d == 2) ? l2
                : (d == 3) ? l3
                : (d == 4) ? l4
                : 0.f;
        row[j] = v;
    }
}

// ---------------------------------------------------------------------------
extern "C" void kernel_launch(void* const* d_in, const int* in_sizes, int n_in,
                              void* d_out, int out_size, void* d_ws, size_t ws_size,
                              hipStream_t stream) {
    (void)in_sizes; (void)n_in; (void)out_size; (void)ws_size;
    const float*         features = (const float*)d_in[0];
    const unsigned char* mask     = (const unsigned char*)d_in[1];
    const int*           dm       = (const int*)d_in[2];
    const float*         tst      = (const float*)d_in[3];
    const float*         codebook = (const float*)d_in[4];
    const float*         W1       = (const float*)d_in[5];
    const float*         b1       = (const float*)d_in[6];
    const float*         W2       = (const float*)d_in[7];
    const float*         b2       = (const float*)d_in[8];

    float* out  = (float*)d_out;
    float* lut  = (float*)d_ws;                       // 7 floats
    float* outX = out;                                // 16*1025*128
    float* outM = outX + (size_t)BB * NP1 * DIN;      // 16*1025
    float* outB = outM + (size_t)BB * NP1;            // 16*1025*1025

    mlp_lut_kernel<<<1, 32, 0, stream>>>(codebook, W1, b1, W2, b2, lut);

    const int xVec = BB * NP1 * (DIN / 4);            // 524800 float4
    build_X_kernel<<<(xVec + 255) / 256, 256, 0, stream>>>(features, tst, outX);

    build_mask_kernel<<<(BB * NP1 + 255) / 256, 256, 0, stream>>>(mask, outM);

    build_bias_kernel<<<BB * NP1, 256, 0, stream>>>(dm, lut, outB);
}